// HollywoodFFT_53584011985637
// MI455X (gfx1250) — compile-verified
//
#include <hip/hip_runtime.h>
#include <hip/hip_bf16.h>

typedef __attribute__((ext_vector_type(2))) float v2f;
typedef __attribute__((ext_vector_type(8))) float v8f;
typedef __attribute__((ext_vector_type(4))) unsigned int u32x4;
typedef __attribute__((ext_vector_type(8))) int i32x8;
typedef __attribute__((ext_vector_type(4))) int i32x4;

#define LSTR 260  // padded LDS row stride (floats): multiple of 4 (16B aligned) and 4-bank shift/row

#if defined(__has_builtin)
#if __has_builtin(__builtin_amdgcn_tensor_load_to_lds) && __has_builtin(__builtin_amdgcn_s_wait_tensorcnt)
#define USE_TDM 1
#endif
#endif

__device__ __forceinline__ v8f wmma4(v2f a, v2f b, v8f c) {
  // D = A(16x4) x B(4x16) + C ; (neg_a, A, neg_b, B, c_mod, C, reuse_a, reuse_b)
  return __builtin_amdgcn_wmma_f32_16x16x4_f32(false, a, false, b, (short)0, c, false, false);
}

#ifdef USE_TDM
// TDM D# for: load a 16x256-float row-tile (one FFT row) from global into LDS,
// hardware-padding each 256-DWORD row by 4 DWORDs -> LDS row stride = 260 floats.
__device__ __forceinline__ void tdm_load_row(const float* gsrc, float* lds_dst) {
  const unsigned long long ga = (unsigned long long)(uintptr_t)gsrc;
  const unsigned int       la = (unsigned int)(uintptr_t)lds_dst;

  u32x4 g0;
  g0.x = 1u;                                    // count=1, is_restore=0, gather off
  g0.y = la;                                    // lds_addr[31:0]
  g0.z = (unsigned int)(ga & 0xFFFFFFFFull);    // global_addr[31:0]
  g0.w = (unsigned int)((ga >> 32) & 0x01FFFFFFull) | (2u << 30); // addr[56:32] | type=2

  i32x8 g1;
  g1[0] = (int)((2u << 16)      // data_size = 4 bytes
              | (1u << 20)      // pad_enable
              | (7u << 22)      // pad_interval: after 256 DWORDs
              | (3u << 25));    // pad_amount: 4 DWORDs (16B) -> stride 260 floats
  g1[1] = (int)(256u << 16);    // tensor_dim0 = 256 (low 16)
  g1[2] = (int)(16u  << 16);    // tensor_dim0 hi = 0 | tensor_dim1 = 16 (low 16)
  g1[3] = (int)(256u << 16);    // tensor_dim1 hi = 0 | tile_dim0 = 256
  g1[4] = (int)16;              // tile_dim1 = 16 | tile_dim2 = 0
  g1[5] = (int)256;             // tensor_dim0_stride = 256 (low 32)
  g1[6] = 0;                    // stride0 hi | tensor_dim1_stride lo (unused)
  g1[7] = 0;

  i32x4 gz  = {0, 0, 0, 0};                     // 2-D tensor: groups 2/3 unused
  i32x8 gz8 = {0, 0, 0, 0, 0, 0, 0, 0};         // extra group (clang-23 6-arg form), unused
  __builtin_amdgcn_tensor_load_to_lds(g0, g1, gz, gz, gz8, 0 /*cpol*/);
}
#endif

__global__ __launch_bounds__(256)
void HollywoodFFT_53584011985637_kernel(const float* __restrict__ xre,
                                        const float* __restrict__ xim,
                                        float* __restrict__ yre,
                                        float* __restrict__ yim) {
  __shared__ __align__(16) float Are[16 * LSTR];
  __shared__ __align__(16) float Aim[16 * LSTR];
  __shared__ __align__(16) float Bre[16 * LSTR];
  __shared__ __align__(16) float Bim[16 * LSTR];

  const int row  = blockIdx.x;
  const int tid  = threadIdx.x;
  const int lane = tid & 31;
  const int wave = tid >> 5;
  const int half = lane >> 4;   // 0: lanes 0-15, 1: lanes 16-31
  const int l15  = lane & 15;

  const float* gre = xre + (size_t)row * 4096;
  const float* gim = xim + (size_t)row * 4096;

  // ---------------- load one row; natural order == stage-1 layout [n2][n0+16*n1] -------------
#ifdef USE_TDM
  if (wave == 0) {              // one wave drives the Tensor Data Mover (EXEC-independent DMA)
    tdm_load_row(gre, Are);
    tdm_load_row(gim, Aim);
  }
#else
#pragma unroll
  for (int j = 0; j < 4; ++j) {
    const int n   = tid * 16 + j * 4;
    const int idx = (n >> 8) * LSTR + (n & 255);
    *(float4*)&Are[idx] = *(const float4*)&gre[n];
    *(float4*)&Aim[idx] = *(const float4*)&gim[n];
  }
#endif

  // ---------------- DFT-16 matrix in WMMA A-operand layout (overlaps the DMA) ----------------
  v2f aRe[4], aIm[4], aImN[4];
#pragma unroll
  for (int kc = 0; kc < 4; ++kc) {
    const int k0 = kc * 4 + half * 2;
    float s, c;
    __sincosf(-0.39269908169872414f * (float)(l15 * k0), &s, &c);      // -2pi/16
    aRe[kc].x = c; aIm[kc].x = s;
    __sincosf(-0.39269908169872414f * (float)(l15 * (k0 + 1)), &s, &c);
    aRe[kc].y = c; aIm[kc].y = s;
    aImN[kc].x = -aIm[kc].x; aImN[kc].y = -aIm[kc].y;
  }

#ifdef USE_TDM
  if (wave == 0) __builtin_amdgcn_s_wait_tensorcnt(0);
#endif
  __syncthreads();

  // ================= stage 1: contract n2 ; read A[r*LSTR+col] ; write B[m][col] =============
#pragma unroll
  for (int t = 0; t < 2; ++t) {
    const int ct = wave + t * 8;          // column tile 0..15 ; ct == n1
    v8f accRe = {}; v8f accIm = {};
#pragma unroll
    for (int kc = 0; kc < 4; ++kc) {
      const int r0 = kc * 4 + half * 2;   // contraction row (n2)
      const int c  = ct * 16 + l15;
      v2f bRe, bIm;
      bRe.x = Are[ r0      * LSTR + c];
      bRe.y = Are[(r0 + 1) * LSTR + c];
      bIm.x = Aim[ r0      * LSTR + c];
      bIm.y = Aim[(r0 + 1) * LSTR + c];
      accRe = wmma4(aRe[kc],  bRe, accRe);
      accRe = wmma4(aImN[kc], bIm, accRe);
      accIm = wmma4(aRe[kc],  bIm, accIm);
      accIm = wmma4(aIm[kc],  bRe, accIm);
    }
    // twiddle W_256^{n1*k0} (n1=ct, k0=m) then store [m][ct*16+l15]
#pragma unroll
    for (int j = 0; j < 8; ++j) {
      const int m = j + 8 * half;
      float s, c;
      __sincosf(-0.02454369260617026f * (float)(m * ct), &s, &c);      // -2pi/256
      const float re = accRe[j], im = accIm[j];
      Bre[m * LSTR + ct * 16 + l15] = re * c - im * s;
      Bim[m * LSTR + ct * 16 + l15] = re * s + im * c;
    }
  }
  __syncthreads();

  // ================= stage 2: contract n1 ; read B[ct*LSTR + l15 + 16*r] ; write A ===========
#pragma unroll
  for (int t = 0; t < 2; ++t) {
    const int ct = wave + t * 8;          // ct == k0
    v8f accRe = {}; v8f accIm = {};
#pragma unroll
    for (int kc = 0; kc < 4; ++kc) {
      const int r0 = kc * 4 + half * 2;   // contraction row (n1)
      v2f bRe, bIm;
      bRe.x = Bre[ct * LSTR + l15 + 16 *  r0     ];
      bRe.y = Bre[ct * LSTR + l15 + 16 * (r0 + 1)];
      bIm.x = Bim[ct * LSTR + l15 + 16 *  r0     ];
      bIm.y = Bim[ct * LSTR + l15 + 16 * (r0 + 1)];
      accRe = wmma4(aRe[kc],  bRe, accRe);
      accRe = wmma4(aImN[kc], bIm, accRe);
      accIm = wmma4(aRe[kc],  bIm, accIm);
      accIm = wmma4(aIm[kc],  bRe, accIm);
    }
    // twiddle W_4096^{n0*(k0+16*k1)} (n0=l15, k0=ct, k1=m) ; store [k1][n0+16*k0]
#pragma unroll
    for (int j = 0; j < 8; ++j) {
      const int m = j + 8 * half;
      float s, c;
      __sincosf(-0.0015339807878856412f * (float)(l15 * (ct + 16 * m)), &s, &c); // -2pi/4096
      const float re = accRe[j], im = accIm[j];
      Are[m * LSTR + ct * 16 + l15] = re * c - im * s;
      Aim[m * LSTR + ct * 16 + l15] = re * s + im * c;
    }
  }
  __syncthreads();

  // ================= stage 3: contract n0 ; read A[ct*LSTR + r + 16*l15] ; store global ======
  float* ore = yre + (size_t)row * 4096;
  float* oim = yim + (size_t)row * 4096;
#pragma unroll
  for (int t = 0; t < 2; ++t) {
    const int ct = wave + t * 8;          // ct == k1
    v8f accRe = {}; v8f accIm = {};
#pragma unroll
    for (int kc = 0; kc < 4; ++kc) {
      const int r0 = kc * 4 + half * 2;   // contraction row (n0)
      v2f bRe, bIm;
      bRe.x = Are[ct * LSTR +  r0      + 16 * l15];
      bRe.y = Are[ct * LSTR + (r0 + 1) + 16 * l15];
      bIm.x = Aim[ct * LSTR +  r0      + 16 * l15];
      bIm.y = Aim[ct * LSTR + (r0 + 1) + 16 * l15];
      accRe = wmma4(aRe[kc],  bRe, accRe);
      accRe = wmma4(aImN[kc], bIm, accRe);
      accIm = wmma4(aRe[kc],  bIm, accIm);
      accIm = wmma4(aIm[kc],  bRe, accIm);
    }
    // output index k = k0 + 16*k1 + 256*k2 = (ct*16 + l15) + 256*m  (digit-reversal folded in)
#pragma unroll
    for (int j = 0; j < 8; ++j) {
      const int m = j + 8 * half;
      const int k = 256 * m + ct * 16 + l15;
      ore[k] = accRe[j];
      oim[k] = accIm[j];
    }
  }
}

extern "C" void kernel_launch(void* const* d_in, const int* in_sizes, int n_in,
                              void* d_out, int out_size, void* d_ws, size_t ws_size,
                              hipStream_t stream) {
  const float* xre = (const float*)d_in[0];
  const float* xim = (const float*)d_in[1];
  float* yre = (float*)d_out;
  float* yim = yre + (size_t)4096 * 4096;
  HollywoodFFT_53584011985637_kernel<<<4096, 256, 0, stream>>>(xre, xim, yre, yim);
}